// Vprime_6682969113355
// MI455X (gfx1250) — compile-verified
//
#include <hip/hip_runtime.h>
#include <cstdint>

// Minmod(theta=2) slope-limited stencil: out = v * delta_rho.
// 8192 rows: rho row = 2048 f32, v/out row = 2044 f32. Pure streaming problem
// (~201 MB, ~8.6 us at 23.3 TB/s). Each block stages a 1028-float rho tile
// into LDS once via gfx1250 async global->LDS B128 (NT hint), then computes
// 4 outputs/thread from LDS with shared half-slopes; v/out move as NT b128.

typedef float v4f __attribute__((ext_vector_type(4)));

#define SLOPE_THETA 2.0f

__device__ __forceinline__ float minmod3(float a, float b, float c) {
  float mn = fminf(fminf(a, b), c);
  float mx = fmaxf(fmaxf(a, b), c);
  // all >=0 -> smallest; all <0 -> largest (closest to 0); mixed -> 0
  return (mn < 0.0f) ? fminf(mx, 0.0f) : mn;
}

__global__ void __launch_bounds__(256) vprime_minmod_kernel(
    const float* __restrict__ rho, const float* __restrict__ v,
    float* __restrict__ out, long long totalRho) {
  constexpr int N    = 2048;   // rho row length
  constexpr int NV   = N - 4;  // outputs per row (2044 = 4*511)
  constexpr int TILE = 1024;   // outputs per block (4 per thread)

  __shared__ __align__(16) float s[TILE + 8];  // 1028 used

  const int bid = blockIdx.x;
  const int row = bid >> 1;            // 2 tiles per row
  const int t0  = (bid & 1) * TILE;    // first output index within row
  const int tid = threadIdx.x;

  // ---- stage rho[row, t0 .. t0+1027] into LDS: 257 async B128 transfers ----
  {
    const long long base = (long long)row * N + t0;
    long long e = base + 4 * (long long)tid;
    if (e > totalRho - 4) e = totalRho - 4;        // clamp tail (data unused)
    uint64_t ga = (uint64_t)(const void*)(rho + e);
    // generic LDS pointer: low 32 bits are the LDS byte address (ISA 10.2)
    uint32_t la = (uint32_t)(uintptr_t)(const void*)(&s[0]) + 16u * (uint32_t)tid;
    asm volatile("global_load_async_to_lds_b128 %0, %1, off th:TH_LOAD_NT"
                 :: "v"(la), "v"(ga) : "memory");
    if (tid == 0) {                                // halo floats 1024..1027
      long long e2 = base + TILE;
      if (e2 > totalRho - 4) e2 = totalRho - 4;
      uint64_t ga2 = (uint64_t)(const void*)(rho + e2);
      uint32_t la2 = (uint32_t)(uintptr_t)(const void*)(&s[0]) + 4u * TILE;
      asm volatile("global_load_async_to_lds_b128 %0, %1, off th:TH_LOAD_NT"
                   :: "v"(la2), "v"(ga2) : "memory");
    }
  }
  asm volatile("s_wait_asynccnt 0" ::: "memory");  // drain this wave's copies
  __syncthreads();                                 // LDS visible block-wide

  const int k = t0 + 4 * tid;                      // first of 4 outputs
  if (k + 3 < NV) {                                // whole-float4 tiles only
    const v4f* s4 = (const v4f*)s;
    const v4f A = s4[tid];
    const v4f B = s4[tid + 1];
    float f[8] = {A[0], A[1], A[2], A[3], B[0], B[1], B[2], B[3]};

    float d[7];
#pragma unroll
    for (int i = 0; i < 7; ++i) d[i] = f[i + 1] - f[i];

    // half_slope[i] = 0.5*minmod(t*d[i], 0.5*(d[i]+d[i+1]), t*d[i+1]); shared
    // across the 4 outputs (6 minmods instead of 12).
    float hs[6];
#pragma unroll
    for (int i = 0; i < 6; ++i)
      hs[i] = 0.5f * minmod3(SLOPE_THETA * d[i], 0.5f * (d[i] + d[i + 1]),
                             SLOPE_THETA * d[i + 1]);

    const long long idx4 = ((long long)row * NV + k) >> 2;  // 16B-aligned
    const v4f vk = __builtin_nontemporal_load((const v4f*)v + idx4);
    v4f o;
#pragma unroll
    for (int j = 0; j < 4; ++j) {
      const float rp = d[j + 1] + (hs[j + 1] - hs[j]);   // result_plus
      const float rm = d[j + 2] - (hs[j + 2] - hs[j + 1]); // result_minus
      const float x = vk[j];
      // jnp.where(v<0, minus, plus): NaN v -> plus branch, matched here
      o[j] = x * ((x < 0.0f) ? rm : rp);
    }
    __builtin_nontemporal_store(o, (v4f*)out + idx4);
  }
}

extern "C" void kernel_launch(void* const* d_in, const int* in_sizes, int n_in,
                              void* d_out, int out_size, void* d_ws, size_t ws_size,
                              hipStream_t stream) {
  const float* rho = (const float*)d_in[0];
  const float* v   = (const float*)d_in[1];
  float* out       = (float*)d_out;
  (void)d_ws; (void)ws_size; (void)n_in; (void)out_size;

  const long long totalRho = (long long)in_sizes[0];
  const int rows = in_sizes[0] / 2048;   // B*M = 8192
  const int tilesPerRow = 2;             // 1024 outputs per block

  dim3 grid((unsigned)(rows * tilesPerRow));
  dim3 block(256);
  vprime_minmod_kernel<<<grid, block, 0, stream>>>(rho, v, out, totalRho);
}